// _spike_lif_30185030157013
// MI455X (gfx1250) — compile-verified
//
#include <hip/hip_runtime.h>

// LIF spiking-neuron forward scan for MI455X (gfx1250).
// Memory-bound: ~839 MB traffic -> ~36us floor at 23.3 TB/s.
// Strategy: async bulk-stage each block's contiguous input region into LDS
// (global_load_async_to_lds_b128, coalesced; single base addr + immediate
// offsets applied to BOTH global and LDS sides per ISA 15.18), then each
// thread scans its own contiguous LDS row; outputs stored as float4.

typedef float v4f __attribute__((ext_vector_type(4)));

#define LIF_ALPHA 0.95f
#define LIF_BETA  0.90f
#define TT   100        // timesteps
#define C4   (TT / 4)   // 25 float4 chunks per element
#define EPB  128        // elements (threads) per block

__global__ __launch_bounds__(EPB)
void lif_scan_kernel(const float* __restrict__ x,
                     float* __restrict__ syn_o,
                     float* __restrict__ mem_o,
                     float* __restrict__ spk_o) {
    // Block's input region: EPB*TT floats, contiguous in global memory AND in
    // the LDS tile (row-major [elem][t]) -> staging is a flat coalesced copy.
    __shared__ v4f tile[EPB * C4];   // 128 * 25 * 16B = 51200 bytes

    const int    tid       = threadIdx.x;
    const size_t blockElem = (size_t)blockIdx.x * EPB;
    const float* gsrc      = x + blockElem * TT;

    // LDS byte offset: shared-aperture flat addresses carry the LDS offset in
    // their low 32 bits (ISA 10.2).
    unsigned   lds_addr = (unsigned)(size_t)(&tile[0]) + (unsigned)tid * 16u;
    const v4f* gbase    = (const v4f*)gsrc + tid;

    // Async global -> LDS bulk copy (ASYNCcnt-tracked). IOFFSET advances both
    // the global source and the LDS destination, so one VGPR address pair and
    // one LDS-address VGPR serve all 25 transfers.
#pragma unroll
    for (int k = 0; k < C4; ++k) {
        asm volatile("global_load_async_to_lds_b128 %0, %1, off offset:%2"
                     :: "v"(lds_addr), "v"(gbase), "i"(k * EPB * 16)
                     : "memory");
    }
    asm volatile("s_wait_asynccnt 0" ::: "memory");
    __syncthreads();

    // Each thread owns one element: its LDS row is contiguous (25 x v4f).
    const v4f*   myrow = tile + (size_t)tid * C4;
    const size_t row   = (blockElem + (size_t)tid) * TT;
    v4f* so = (v4f*)(syn_o + row);
    v4f* mo = (v4f*)(mem_o + row);
    v4f* ko = (v4f*)(spk_o + row);

    float syn = 0.0f, mem = 0.0f;

    // One reference scan step: spike from OLD mem, mem update uses OLD syn;
    // emitted values are (syn_next, mem_next, spike).
#define LIF_STEP(XC, SC, MC, OC)                              \
    {                                                         \
        float ot = ((mem - 1.0f) > 0.0f) ? 1.0f : 0.0f;       \
        float sn = LIF_ALPHA * syn + (XC);                    \
        float mn = (LIF_BETA * mem + syn) * (1.0f - ot);      \
        (OC) = ot; (SC) = sn; (MC) = mn;                      \
        syn = sn; mem = mn;                                   \
    }

#pragma unroll 5
    for (int c = 0; c < C4; ++c) {
        v4f xv = myrow[c];          // ds_load_b128, conflict-light (stride 100)
        v4f s, m, o;
        LIF_STEP(xv.x, s.x, m.x, o.x);
        LIF_STEP(xv.y, s.y, m.y, o.y);
        LIF_STEP(xv.z, s.z, m.z, o.z);
        LIF_STEP(xv.w, s.w, m.w, o.w);
        so[c] = s;                  // global_store_b128; consecutive c's from
        mo[c] = m;                  // this thread fill full cachelines in L2
        ko[c] = o;
    }
#undef LIF_STEP
}

extern "C" void kernel_launch(void* const* d_in, const int* in_sizes, int n_in,
                              void* d_out, int out_size, void* d_ws, size_t ws_size,
                              hipStream_t stream) {
    const float* x   = (const float*)d_in[0];
    float*       out = (float*)d_out;

    const size_t NT = (size_t)in_sizes[0];   // B*C*H*W*T = 52,428,800
    const size_t N  = NT / TT;               // 524,288 elements

    float* syn_o = out;
    float* mem_o = out + NT;
    float* spk_o = out + 2 * NT;

    dim3 grid((unsigned)(N / EPB));          // 4096 blocks
    dim3 block(EPB);                         // 128 threads = 4 wave32
    lif_scan_kernel<<<grid, block, 0, stream>>>(x, syn_o, mem_o, spk_o);
}